// UnpadBmm2_3848290697285
// MI455X (gfx1250) — compile-verified
//
#include <hip/hip_runtime.h>

typedef __attribute__((ext_vector_type(2))) float v2f;
typedef __attribute__((ext_vector_type(8))) float v8f;

#define HEADS 16
#define HIDDEN 1024
#define HS 64
#define WAVES_PER_BLOCK 8

// One wave computes a 16x64 output block: (seq i, head h, 16-row q-tile) x all
// four 16-col d-tiles of that head. K-loop steps by 4 using
// V_WMMA_F32_16X16X4_F32; the A fragment (float2/lane) is loaded once per step
// and reused by 4 WMMAs so the 723MB A tensor is streamed from HBM exactly once.
// A loads / C stores are non-temporal so the 36MB B tensor (re-read s/16 times)
// stays resident in the 192MB L2.
__global__ __launch_bounds__(256) void ragged_bmm2_wmma_kernel(
    const float* __restrict__ a, const float* __restrict__ b,
    const int* __restrict__ seqlen, int n_seq,
    float* __restrict__ out, int total_qt)
{
    const int lane = threadIdx.x & 31;
    // Force wave id / q-tile id scalar so the tile lookup and the K-loop are
    // pure SALU/SMEM: the loop exit becomes a scalar branch and EXEC stays
    // all-1s around the WMMAs (ISA requirement).
    const int wave = __builtin_amdgcn_readfirstlane((int)threadIdx.x) >> 5;
    const int qt   = __builtin_amdgcn_readfirstlane((int)blockIdx.x) * WAVES_PER_BLOCK + wave;
    if (qt >= total_qt) return;          // scalar branch
    const int head = __builtin_amdgcn_readfirstlane((int)blockIdx.y);

    // Locate the sequence containing this q-tile (ragged prefix scan, <=8 iters,
    // all scalar: uniform pointer + uniform index -> SMEM loads).
    long a_off = 0;        // element offset of seq's A block in packed 'a'
    long t_off = 0;        // token (row) offset of seq in 'b' / 'out'
    int  qt_base = 0;
    int  s = 0;
    for (int i = 0; i < n_seq; ++i) {
        const int si = seqlen[i];
        const int nq = si >> 4;          // q-tiles in this seq
        if (qt < qt_base + nq) { s = si; break; }
        qt_base += nq;
        a_off   += (long)HEADS * si * si;
        t_off   += si;
    }
    const int qt_local = qt - qt_base;

    // All seqlens are multiples of 16 -> trip count s/4 is a multiple of 4:
    // lets the 4x unroll drop its remainder loop entirely.
    __builtin_assume(s > 0);
    __builtin_assume((s & 15) == 0);

    // WMMA f32 16x16x4 lane mapping (ISA 7.12.2):
    //  A: lane m=lane&15 holds K = 2*half + {0,1}  (half = lane>>4)
    //  B: lane n=lane&15 holds rows K = 2*half + {0,1}
    //  C/D: VGPR r -> row r + 8*half, col n
    const int half = lane >> 4;
    const int mn   = lane & 15;

    const float* Arow = a + a_off + (long)head * s * s
                          + (long)(qt_local * 16 + mn) * s;      // A row m
    const float* Bcol = b + (long)t_off * HIDDEN + head * HS + mn;
    float*       Crow = out + ((long)t_off + qt_local * 16) * HIDDEN + head * HS;

    v8f acc0 = {}, acc1 = {}, acc2 = {}, acc3 = {};

    #pragma unroll 4
    for (int k = 0; k < s; k += 4) {
        const int kk = k + 2 * half;
        // A fragment: contiguous 8B per lane; non-temporal (single-use stream)
        v2f afrag = __builtin_nontemporal_load((const v2f*)(Arow + kk));
        // B fragments: two rows (kk, kk+1), 16 consecutive cols per half-wave;
        // regular-temporal so B stays hot in L2 across q-tiles.
        const float* b0p = Bcol + (size_t)kk * HIDDEN;
        const float* b1p = b0p + HIDDEN;
        v2f bf0 = { b0p[0],  b1p[0]  };
        v2f bf1 = { b0p[16], b1p[16] };
        v2f bf2 = { b0p[32], b1p[32] };
        v2f bf3 = { b0p[48], b1p[48] };

        acc0 = __builtin_amdgcn_wmma_f32_16x16x4_f32(false, afrag, false, bf0,
                                                     (short)0, acc0, false, false);
        acc1 = __builtin_amdgcn_wmma_f32_16x16x4_f32(false, afrag, false, bf1,
                                                     (short)0, acc1, false, false);
        acc2 = __builtin_amdgcn_wmma_f32_16x16x4_f32(false, afrag, false, bf2,
                                                     (short)0, acc2, false, false);
        acc3 = __builtin_amdgcn_wmma_f32_16x16x4_f32(false, afrag, false, bf3,
                                                     (short)0, acc3, false, false);
    }

    // Store 16x64 block, non-temporal (no reuse). VGPR r holds row (r + 8*half),
    // col mn (+ 16*dt).
    #pragma unroll
    for (int r = 0; r < 8; ++r) {
        float* crow = Crow + (size_t)(r + 8 * half) * HIDDEN + mn;
        __builtin_nontemporal_store(acc0[r], crow + 0);
        __builtin_nontemporal_store(acc1[r], crow + 16);
        __builtin_nontemporal_store(acc2[r], crow + 32);
        __builtin_nontemporal_store(acc3[r], crow + 48);
    }
}

extern "C" void kernel_launch(void* const* d_in, const int* in_sizes, int n_in,
                              void* d_out, int out_size, void* d_ws, size_t ws_size,
                              hipStream_t stream) {
    const float* a      = (const float*)d_in[0];
    const float* b      = (const float*)d_in[1];
    const int*   seqlen = (const int*)d_in[2];
    const int    n_seq  = in_sizes[2];

    // hidden=1024 per reference; total tokens = in_sizes[1] / hidden
    const int total_tokens = in_sizes[1] / HIDDEN;
    const int total_qt     = total_tokens / 16;   // all seqlens are multiples of 16

    dim3 grid((total_qt + WAVES_PER_BLOCK - 1) / WAVES_PER_BLOCK, HEADS, 1);
    ragged_bmm2_wmma_kernel<<<grid, 256, 0, stream>>>(
        a, b, seqlen, n_seq, (float*)d_out, total_qt);
}